// LGGNet_55825984914104
// MI455X (gfx1250) — compile-verified
//
#include <hip/hip_runtime.h>

#define BB 64
#define NN 62
#define DD 4
#define TT 512
#define NP 64          // N padded to 64 for 16x16 tiling
#define EPSF 1e-5f

typedef float v2f __attribute__((ext_vector_type(2)));
typedef float v8f __attribute__((ext_vector_type(8)));

__global__ __launch_bounds__(256) void lggnet_kernel(
    const float* __restrict__ x,    // (B,N,D,T)
    const float* __restrict__ lw,   // (N,D)
    const float* __restrict__ lb,   // (1,N,1)
    const float* __restrict__ ga,   // (N,N)
    const float* __restrict__ gw,   // (D,D)
    const float* __restrict__ gb,   // (1,1,D)
    const float* __restrict__ g1, const float* __restrict__ b1,
    const float* __restrict__ g2, const float* __restrict__ b2,
    float* __restrict__ out)        // (B,N,D,T)
{
  __shared__ float gsym[NP * NP];   // symmetrized global adj, zero padded
  __shared__ float adj[NP * NP];    // per-b un-normalized adjacency
  __shared__ float outb[NP * DD];   // per-b local-filter output (padded rows 0)
  __shared__ float hd[NP * DD];     // d_j * (BN1(out) @ gcn_w - gcn_b)
  __shared__ float scale1[NP], shift1[NP], dvec[NP];
  __shared__ float scale2[NP], shift2[NP];
  __shared__ float red[256], red2[256];

  const int t    = blockIdx.x;
  const int tid  = threadIdx.x;
  const int n    = tid >> 2;            // node index for (n,d) threads
  const int d    = tid & 3;             // feature index
  const int lane = tid & 31;
  const int wave = tid >> 5;
  const int mRow = lane & 15;           // WMMA A/B: lane -> M (or N)
  const int kb   = (lane >> 4) << 1;    // K pair base: 0 or 2

  // ---- symmetrized adjacency (shared by all t,b) ----
  for (int i = tid; i < NP * NP; i += 256) {
    int r = i >> 6, c = i & 63;
    gsym[i] = (r < NN && c < NN) ? (ga[r * NN + c] + ga[c * NN + r]) : 0.f;
  }

  // ---- BN1 stats: per (t,n) over (b,d) of out = relu(x*w - b) ----
  float s1 = 0.f, q1 = 0.f;
  if (n < NN) {
    const float w   = lw[n * DD + d];
    const float bia = lb[n];
    const float* xp = x + (size_t)(n * DD + d) * TT + t;
    for (int b = 0; b < BB; ++b) {
      float o = fmaxf(xp[(size_t)b * (NN * DD * TT)] * w - bia, 0.f);
      s1 += o; q1 += o * o;
    }
  }
  red[tid] = s1; red2[tid] = q1;
  __syncthreads();
  if (tid < NN) {
    float su = red [4*tid] + red [4*tid+1] + red [4*tid+2] + red [4*tid+3];
    float sq = red2[4*tid] + red2[4*tid+1] + red2[4*tid+2] + red2[4*tid+3];
    float mean = su * (1.f / (BB * DD));
    float var  = sq * (1.f / (BB * DD)) - mean * mean;  // biased (training)
    float sc   = g1[tid] * rsqrtf(var + EPSF);
    scale1[tid] = sc;
    shift1[tid] = b1[tid] - mean * sc;
  }
  __syncthreads();

  float ys = 0.f, yq = 0.f;           // BN2 running stats per (n,d) thread
  for (int b = 0; b < BB; ++b) {
    // local filter for this (t,b); padded rows (n>=62) -> 0
    float o = 0.f;
    if (n < NN) {
      float xv = x[(((size_t)b * NN + n) * DD + d) * TT + t];
      o = fmaxf(xv * lw[n * DD + d] - lb[n], 0.f);
    }
    outb[tid] = o;
    __syncthreads();

    // s = O @ O^T via V_WMMA_F32_16X16X4_F32 (K = D = 4 exactly);
    // 16 tiles, 2 per wave.  adj = relu(s .* gsym) + I
#pragma unroll
    for (int k = 0; k < 2; ++k) {
      const int tIdx = wave * 2 + k;
      const int r = tIdx >> 2, c = tIdx & 3;
      v2f av = *(const v2f*)(&outb[(16 * r + mRow) * DD + kb]);
      v2f bv = *(const v2f*)(&outb[(16 * c + mRow) * DD + kb]);
      v8f cc = {0.f, 0.f, 0.f, 0.f, 0.f, 0.f, 0.f, 0.f};
      cc = __builtin_amdgcn_wmma_f32_16x16x4_f32(false, av, false, bv,
                                                 (short)0, cc, false, false);
      const int iBase = 16 * r + ((lane >> 4) << 3);   // C layout: M = v (+8)
      const int j     = 16 * c + mRow;                 // N = lane%16
#pragma unroll
      for (int v = 0; v < 8; ++v) {
        int i = iBase + v;
        float val = fmaxf(cc[v] * gsym[i * NP + j], 0.f);
        if (i == j) val += 1.f;
        adj[i * NP + j] = val;
      }
    }
    __syncthreads();

    // degree: d_i = rsqrt(rowsum) with zero guard
    if (tid < NP) {
      float rs = 0.f;
      for (int j = 0; j < NP; ++j) rs += adj[tid * NP + j];
      if (rs == 0.f) rs = 1.f;
      dvec[tid] = rsqrtf(rs);
    }
    __syncthreads();

    // hd[j][:] = d_j * (BN1(out)[j] @ gcn_w - gcn_b)
    float hv = 0.f;
    if (n < NN) {
      float sc = scale1[n], sh = shift1[n];
      hv = (sc * outb[n * 4 + 0] + sh) * gw[0 * 4 + d]
         + (sc * outb[n * 4 + 1] + sh) * gw[1 * 4 + d]
         + (sc * outb[n * 4 + 2] + sh) * gw[2 * 4 + d]
         + (sc * outb[n * 4 + 3] + sh) * gw[3 * 4 + d]
         - gb[d];
      hv *= dvec[n];
    }
    hd[tid] = hv;
    __syncthreads();

    // y = relu(d_i * (A @ hd)); stage y in-place at its final output slot
    // (written and later re-read by the SAME thread -> no cross-thread hazard)
    float y = 0.f;
    if (n < NN) {
      float acc = 0.f;
      for (int j = 0; j < NP; ++j) acc += adj[n * NP + j] * hd[j * DD + d];
      y = fmaxf(dvec[n] * acc, 0.f);
      out[(((size_t)b * NN + n) * DD + d) * TT + t] = y;
    }
    ys += y; yq += y * y;
    __syncthreads();
  }

  // ---- BN2 ----
  red[tid] = ys; red2[tid] = yq;
  __syncthreads();
  if (tid < NN) {
    float su = red [4*tid] + red [4*tid+1] + red [4*tid+2] + red [4*tid+3];
    float sq = red2[4*tid] + red2[4*tid+1] + red2[4*tid+2] + red2[4*tid+3];
    float mean = su * (1.f / (BB * DD));
    float var  = sq * (1.f / (BB * DD)) - mean * mean;
    float sc   = g2[tid] * rsqrtf(var + EPSF);
    scale2[tid] = sc;
    shift2[tid] = b2[tid] - mean * sc;
  }
  __syncthreads();

  // apply BN2 affine to the staged y values (same-thread read-modify-write)
  if (n < NN) {
    float sc = scale2[n], sh = shift2[n];
    for (int b = 0; b < BB; ++b) {
      size_t idx = (((size_t)b * NN + n) * DD + d) * TT + t;
      out[idx] = sc * out[idx] + sh;
    }
  }
}

extern "C" void kernel_launch(void* const* d_in, const int* in_sizes, int n_in,
                              void* d_out, int out_size, void* d_ws, size_t ws_size,
                              hipStream_t stream) {
  (void)in_sizes; (void)n_in; (void)d_ws; (void)ws_size; (void)out_size;
  const float* x  = (const float*)d_in[0];
  const float* lw = (const float*)d_in[1];
  const float* lb = (const float*)d_in[2];
  const float* ga = (const float*)d_in[3];
  const float* gw = (const float*)d_in[4];
  const float* gb = (const float*)d_in[5];
  const float* g1 = (const float*)d_in[6];
  const float* b1 = (const float*)d_in[7];
  const float* g2 = (const float*)d_in[8];
  const float* b2 = (const float*)d_in[9];
  lggnet_kernel<<<TT, 256, 0, stream>>>(x, lw, lb, ga, gw, gb,
                                        g1, b1, g2, b2, (float*)d_out);
}